// SinglePassReadModel_3204045603482
// MI455X (gfx1250) — compile-verified
//
#include <hip/hip_runtime.h>
#include <hip/hip_bf16.h>

#define HIDDEN 64
#define VOCAB  64
#define KSEL   8
#define SEQLEN 2048

typedef __attribute__((ext_vector_type(2))) float v2f;
typedef __attribute__((ext_vector_type(8))) float v8f;

// ---------------------------------------------------------------------------
// Kernel A: one workgroup. Precompute per-vocab encoder table H_table (64x64),
// query table Q_table (64x64), gate scores s_table (64), and token order
// sorted by descending score (stable: lowest index wins ties).
// ---------------------------------------------------------------------------
__global__ __launch_bounds__(256) void precompute_tables(
    const float* __restrict__ embed, const float* __restrict__ W1,
    const float* __restrict__ b1,    const float* __restrict__ W2,
    const float* __restrict__ b2,    const float* __restrict__ ln_g,
    const float* __restrict__ ln_b,  const float* __restrict__ gate_w,
    const float* __restrict__ gate_b,const float* __restrict__ Wq,
    const float* __restrict__ bq,
    float* __restrict__ H_table, float* __restrict__ Q_table,
    float* __restrict__ s_table, int* __restrict__ order) {
  __shared__ float sE[VOCAB * HIDDEN];        // 16 KB
  __shared__ float sT1[VOCAB * 2 * HIDDEN];   // 32 KB
  __shared__ float sX[VOCAB * HIDDEN];        // 16 KB
  __shared__ float sS[VOCAB];

  const int tid = threadIdx.x;

  for (int i = tid; i < VOCAB * HIDDEN; i += 256) sE[i] = embed[i];
  __syncthreads();

  // T1 = relu(E @ W1 + b1) : 64 x 128
  for (int i = tid; i < VOCAB * 2 * HIDDEN; i += 256) {
    const int r = i >> 7, j = i & 127;
    float s = b1[j];
    for (int k = 0; k < HIDDEN; ++k) s += sE[r * HIDDEN + k] * W1[k * 2 * HIDDEN + j];
    sT1[i] = fmaxf(s, 0.0f);
  }
  __syncthreads();

  // X = E + T1 @ W2 + b2 : 64 x 64
  for (int i = tid; i < VOCAB * HIDDEN; i += 256) {
    const int r = i >> 6, c = i & 63;
    float s = b2[c];
    for (int k = 0; k < 2 * HIDDEN; ++k) s += sT1[r * 2 * HIDDEN + k] * W2[k * HIDDEN + c];
    sX[i] = sE[i] + s;
  }
  __syncthreads();

  // LayerNorm each row (threads 0..63, one row each)
  if (tid < VOCAB) {
    float mu = 0.0f;
    for (int c = 0; c < HIDDEN; ++c) mu += sX[tid * HIDDEN + c];
    mu *= (1.0f / HIDDEN);
    float var = 0.0f;
    for (int c = 0; c < HIDDEN; ++c) {
      const float d = sX[tid * HIDDEN + c] - mu;
      var += d * d;
    }
    var *= (1.0f / HIDDEN);
    const float inv = rsqrtf(var + 1e-5f);
    for (int c = 0; c < HIDDEN; ++c)
      sX[tid * HIDDEN + c] = (sX[tid * HIDDEN + c] - mu) * inv * ln_g[c] + ln_b[c];
  }
  __syncthreads();

  // Export H_table; gate scores per vocab id
  for (int i = tid; i < VOCAB * HIDDEN; i += 256) H_table[i] = sX[i];
  if (tid < VOCAB) {
    float s = gate_b[0];
    for (int c = 0; c < HIDDEN; ++c) s += sX[tid * HIDDEN + c] * gate_w[c];
    sS[tid] = s;
    s_table[tid] = s;
  }
  __syncthreads();

  // Stable descending sort of token ids by score (selection, 64 entries)
  if (tid == 0) {
    unsigned long long used = 0ull;
    for (int i = 0; i < VOCAB; ++i) {
      int best = 0;
      float bs = -3.4e38f;
      for (int t = 0; t < VOCAB; ++t) {
        if (!((used >> t) & 1ull) && sS[t] > bs) { bs = sS[t]; best = t; }
      }
      used |= (1ull << best);
      order[i] = best;
    }
  }

  // Q_table = H_table @ Wq + bq : 64 x 64
  for (int i = tid; i < VOCAB * HIDDEN; i += 256) {
    const int r = i >> 6, j = i & 63;
    float s = bq[j];
    for (int k = 0; k < HIDDEN; ++k) s += sX[r * HIDDEN + k] * Wq[k * HIDDEN + j];
    Q_table[i] = s;
  }
}

// ---------------------------------------------------------------------------
// Kernel B: one workgroup per batch row. LDS histogram of token ids over the
// full sequence (the ONLY O(B*L) data touch: 2 MB total), then top-8 selection
// via the precomputed score order, tiny 8-slot attention -> ctx (B x 64).
// ---------------------------------------------------------------------------
__global__ __launch_bounds__(256) void select_and_attend(
    const int* __restrict__ seq,
    const float* __restrict__ H_table, const float* __restrict__ Q_table,
    const int* __restrict__ order,
    float* __restrict__ ctx, int L) {
  __shared__ int   cnt[VOCAB];
  __shared__ int   sel[KSEL];
  __shared__ float logit[KSEL];
  __shared__ float attn[KSEL];
  __shared__ float q[HIDDEN];

  const int b   = blockIdx.x;
  const int tid = threadIdx.x;
  const int* row = seq + (size_t)b * L;

  if (tid < VOCAB) cnt[tid] = 0;
  __syncthreads();

  for (int l = tid; l < L; l += 256) atomicAdd(&cnt[row[l] & (VOCAB - 1)], 1);

  if (tid < HIDDEN) q[tid] = Q_table[(row[L - 1] & (VOCAB - 1)) * HIDDEN + tid];
  __syncthreads();

  if (tid == 0) {
    int rem = KSEL, p = 0;
    for (int i = 0; i < VOCAB && rem > 0; ++i) {
      const int t = order[i];
      int c = cnt[t];
      if (c > rem) c = rem;
      for (int j = 0; j < c; ++j) sel[p++] = t;
      rem -= c;
    }
  }
  __syncthreads();

  if (tid < KSEL) {
    const float* hv = H_table + sel[tid] * HIDDEN;
    float s = 0.0f;
    for (int h = 0; h < HIDDEN; ++h) s += hv[h] * q[h];
    logit[tid] = s * 0.125f;  // 1/sqrt(64)
  }
  __syncthreads();

  if (tid == 0) {
    float mx = logit[0];
    for (int k = 1; k < KSEL; ++k) mx = fmaxf(mx, logit[k]);
    float den = 0.0f;
    for (int k = 0; k < KSEL; ++k) { attn[k] = expf(logit[k] - mx); den += attn[k]; }
    const float inv = 1.0f / den;
    for (int k = 0; k < KSEL; ++k) attn[k] *= inv;
  }
  __syncthreads();

  if (tid < HIDDEN) {
    float s = 0.0f;
    for (int k = 0; k < KSEL; ++k) s += attn[k] * H_table[sel[k] * HIDDEN + tid];
    ctx[b * HIDDEN + tid] = s;
  }
}

// ---------------------------------------------------------------------------
// Kernel C: out = ctx (B x 64) @ Wo (64 x 64) + bo, fp32 via
// V_WMMA_F32_16X16X4_F32. One wave32 per 16x16 output tile; 16 K-steps of 4.
// A (16x4 f32): lanes 0-15 hold K=kk+0,kk+1; lanes 16-31 hold K=kk+2,kk+3.
// D (16x16 f32): reg r = row r (lanes 0-15) / row r+8 (lanes 16-31), N = lane&15.
// ---------------------------------------------------------------------------
__global__ __launch_bounds__(32) void gemm_ctx_wo(
    const float* __restrict__ ctx, const float* __restrict__ Wo,
    const float* __restrict__ bo, float* __restrict__ out) {
  const int tile = blockIdx.x;
  const int tm = tile >> 2;   // M tile (16 rows each)
  const int tn = tile & 3;    // N tile (16 cols each)
  const int lane = threadIdx.x;
  const int half = lane >> 4;
  const int l16  = lane & 15;
  const int arow = tm * 16 + l16;  // A row this lane supplies
  const int bcol = tn * 16 + l16;  // B col this lane supplies

  v8f acc = {};
#pragma unroll
  for (int kk = 0; kk < HIDDEN; kk += 4) {
    const int k0 = kk + 2 * half;
    v2f a, bm;
    a.x  = ctx[arow * HIDDEN + k0 + 0];
    a.y  = ctx[arow * HIDDEN + k0 + 1];
    bm.x = Wo[(k0 + 0) * VOCAB + bcol];
    bm.y = Wo[(k0 + 1) * VOCAB + bcol];
    acc = __builtin_amdgcn_wmma_f32_16x16x4_f32(
        /*neg_a=*/false, a, /*neg_b=*/false, bm,
        /*c_mod=*/(short)0, acc, /*reuse_a=*/false, /*reuse_b=*/false);
  }

  const float bias = bo[bcol];
#pragma unroll
  for (int r = 0; r < 8; ++r) {
    const int m = tm * 16 + r + half * 8;
    out[m * VOCAB + bcol] = acc[r] + bias;
  }
}

// ---------------------------------------------------------------------------
extern "C" void kernel_launch(void* const* d_in, const int* in_sizes, int n_in,
                              void* d_out, int out_size, void* d_ws, size_t ws_size,
                              hipStream_t stream) {
  const int* seq      = (const int*)d_in[0];
  const float* embed  = (const float*)d_in[1];
  const float* W1     = (const float*)d_in[2];
  const float* b1     = (const float*)d_in[3];
  const float* W2     = (const float*)d_in[4];
  const float* b2     = (const float*)d_in[5];
  const float* ln_g   = (const float*)d_in[6];
  const float* ln_b   = (const float*)d_in[7];
  const float* gate_w = (const float*)d_in[8];
  const float* gate_b = (const float*)d_in[9];
  const float* Wq     = (const float*)d_in[10];
  const float* bq     = (const float*)d_in[11];
  const float* Wo     = (const float*)d_in[12];
  const float* bo     = (const float*)d_in[13];
  float* out = (float*)d_out;

  const int L = SEQLEN;
  const int B = in_sizes[0] / L;

  // Workspace layout (fp32 words)
  float* H_table = (float*)d_ws;           // 64*64
  float* Q_table = H_table + VOCAB * HIDDEN;  // 64*64
  float* s_table = Q_table + VOCAB * HIDDEN;  // 64
  int*   order   = (int*)(s_table + VOCAB);   // 64
  float* ctx     = (float*)(order + VOCAB);   // B*64

  precompute_tables<<<1, 256, 0, stream>>>(embed, W1, b1, W2, b2, ln_g, ln_b,
                                           gate_w, gate_b, Wq, bq,
                                           H_table, Q_table, s_table, order);

  select_and_attend<<<B, 256, 0, stream>>>(seq, H_table, Q_table, order, ctx, L);

  const int tiles = (B / 16) * (VOCAB / 16);
  gemm_ctx_wo<<<tiles, 32, 0, stream>>>(ctx, Wo, bo, out);
}